// JarvenClassifier_1322849927506
// MI455X (gfx1250) — compile-verified
//
#include <hip/hip_runtime.h>
#include <hip/hip_bf16.h>
#include <math.h>

// ---------------- constants ----------------
static constexpr int Bn   = 8;
static constexpr int Sn   = 512;
static constexpr int Dn   = 768;
static constexpr int Tn   = 16;
static constexpr int LTn  = 4;
static constexpr int Cn   = 8;
static constexpr int LCn  = 6;
static constexpr int NCn  = 128;    // T*C
static constexpr int Hn   = 512;    // LSTM hidden
static constexpr int Gn   = 2048;   // 4*H
static constexpr int En   = 1024;   // EMB
static constexpr int DHn  = 128;    // head dim
static constexpr int NH   = 8;      // heads

typedef float v2f __attribute__((ext_vector_type(2)));
typedef float v8f __attribute__((ext_vector_type(8)));

// ---------------- WMMA fp32 GEMM with async-LDS weight staging ----------------
// C[M,N] = act( A[M,K] @ W[N,K]^T + bias[N] + Z[M,N] )
// grid = (N/64, M/128), block = 128 (4 waves). Block tile 128M x 64N.
// Each wave: 32M x 64N via 2 A-frags x 4 B-frags -> 8 f32 accumulators.
// W[n0:n0+64, kc:kc+32] is staged once per block into LDS with
// global_load_async_to_lds_b128 (ASYNCcnt), then read as B-frags via ds_load_b64.
__global__ __launch_bounds__(128)
void wmma_gemm(const float* __restrict__ A, int lda,
               const float* __restrict__ W, int ldw,
               const float* __restrict__ bias,
               const float* __restrict__ Z, long long ldz,
               float* __restrict__ C, int ldc,
               int K, int leaky)
{
    // 64 N-rows x 32 K, padded to 36-float stride (144B: 16B-aligned rows,
    // conflict-free ds_load_b64 B-frag reads)
    __shared__ __align__(16) float tileW[64][36];

    const int tid  = threadIdx.x;
    const int wave = tid >> 5;
    const int lane = tid & 31;
    const int half = lane >> 4;     // 0: K+0/1, 1: K+2/3 (A/B frag layout)
    const int l16  = lane & 15;
    const int m0 = blockIdx.y * 128 + wave * 32;
    const int n0 = blockIdx.x * 64;

    const float* arow0 = A + (size_t)(m0 + l16) * lda;
    const float* arow1 = A + (size_t)(m0 + 16 + l16) * lda;
    const float* wbase = W + (size_t)n0 * ldw;

    v8f acc[2][4] = {};

    for (int kc = 0; kc < K; kc += 32) {
        // ---- async-stage W chunk into LDS (4 b128 issues per thread, uniform) ----
        for (int i = tid; i < 512; i += 128) {
            const int row = i >> 3;          // 0..63
            const int g   = i & 7;           // 16B group within the 32-K chunk
            unsigned lds_off = (unsigned)(size_t)&tileW[row][g * 4];
            unsigned long long gaddr =
                (unsigned long long)(size_t)(wbase + (size_t)row * ldw + kc + g * 4);
            asm volatile("global_load_async_to_lds_b128 %0, %1, off"
                         :: "v"(lds_off), "v"(gaddr) : "memory");
        }
        asm volatile("s_wait_asynccnt 0x0" ::: "memory");
        __syncthreads();

#pragma unroll 2
        for (int k = 0; k < 32; k += 4) {
            const int kk = k + 2 * half;
            v2f a0; a0.x = arow0[kc + kk]; a0.y = arow0[kc + kk + 1];
            v2f a1; a1.x = arow1[kc + kk]; a1.y = arow1[kc + kk + 1];
            v2f b[4];
#pragma unroll
            for (int t = 0; t < 4; ++t)
                b[t] = *(const v2f*)&tileW[t * 16 + l16][kk];
#pragma unroll
            for (int t = 0; t < 4; ++t) {
                acc[0][t] = __builtin_amdgcn_wmma_f32_16x16x4_f32(
                    false, a0, false, b[t], (short)0, acc[0][t], false, false);
                acc[1][t] = __builtin_amdgcn_wmma_f32_16x16x4_f32(
                    false, a1, false, b[t], (short)0, acc[1][t], false, false);
            }
        }
        __syncthreads();
    }

    // D layout: VGPR r holds row M = base + r + 8*half, col N = ntile*16 + l16
#pragma unroll
    for (int aIdx = 0; aIdx < 2; ++aIdx) {
#pragma unroll
        for (int t = 0; t < 4; ++t) {
            const int n = n0 + t * 16 + l16;
            const float bv = bias ? bias[n] : 0.0f;
#pragma unroll
            for (int r = 0; r < 8; ++r) {
                const int m = m0 + aIdx * 16 + r + 8 * half;
                float v = acc[aIdx][t][r] + bv;
                if (Z) v += Z[(size_t)m * ldz + n];
                if (leaky) v = v > 0.0f ? v : 0.01f * v;
                C[(size_t)m * ldc + n] = v;
            }
        }
    }
}

// ---------------- gather span rows ----------------
// out[row, :] = seq[b = row / rows_per_batch][ids[row]][:] ; one block per row
__global__ void gather_rows(const float* __restrict__ seq, const int* __restrict__ ids,
                            float* __restrict__ out, int rows_per_batch)
{
    const int row = blockIdx.x;
    const int b = row / rows_per_batch;
    const int id = ids[row];
    const float4* src = (const float4*)(seq + ((size_t)b * Sn + id) * Dn);
    float4* dst = (float4*)(out + (size_t)row * Dn);
    for (int i = threadIdx.x; i < Dn / 4; i += blockDim.x) dst[i] = src[i];
}

// ---------------- bias fold: o = a + b ----------------
__global__ void vec_add(const float* __restrict__ a, const float* __restrict__ b,
                        float* __restrict__ o, int n)
{
    const int i = blockIdx.x * blockDim.x + threadIdx.x;
    if (i < n) o[i] = a[i] + b[i];
}

// ---------------- LSTM gate nonlinearity ----------------
// G: [Ns, 4H] preactivations (i,f,g,o blocks). Updates c,h; optionally scatters
// h into the layer-1 input buffer at stride hs_stride.
__global__ void lstm_gate(const float* __restrict__ G, float* __restrict__ Cst,
                          float* __restrict__ Hst, float* __restrict__ hs,
                          long long hs_stride, int Ns)
{
    const int idx = blockIdx.x * blockDim.x + threadIdx.x;
    if (idx >= Ns * Hn) return;
    const int s = idx / Hn, j = idx % Hn;
    const float* g = G + (size_t)s * Gn;
    const float gi = g[j];
    const float gf = g[Hn + j];
    const float gg = g[2 * Hn + j];
    const float go = g[3 * Hn + j];
    const float i_ = 1.0f / (1.0f + expf(-gi));
    const float f_ = 1.0f / (1.0f + expf(-gf));
    const float o_ = 1.0f / (1.0f + expf(-go));
    const float c = f_ * Cst[idx] + i_ * tanhf(gg);
    const float h = o_ * tanhf(c);
    Cst[idx] = c;
    Hst[idx] = h;
    if (hs) hs[(size_t)s * hs_stride + j] = h;
}

// ---------------- copy [Ns,H] into strided columns of [Ns,E] ----------------
__global__ void copy_cols(const float* __restrict__ src, float* __restrict__ dst,
                          int n, int dst_stride)
{
    const int idx = blockIdx.x * blockDim.x + threadIdx.x;
    if (idx >= n) return;
    const int s = idx / Hn, j = idx % Hn;
    dst[(size_t)s * dst_stride + j] = src[idx];
}

// ---------------- cross attention: 1 query x 8 keys per (b,t,head) ----------------
// q: [B*T, E], k/v: [B*T*C, E]; one wave per (bt, head); dh=128 -> 4 floats/lane.
__global__ __launch_bounds__(32)
void mha_attn(const float* __restrict__ q, const float* __restrict__ k,
              const float* __restrict__ v, float* __restrict__ out)
{
    const int gid  = blockIdx.x;
    const int head = gid % NH;
    const int bt   = gid / NH;
    const int lane = threadIdx.x;
    const size_t hoff = (size_t)head * DHn + (size_t)lane * 4;

    const float4 qv = *(const float4*)(q + (size_t)bt * En + hoff);
    const float scale = 0.08838834764831845f;  // 1/sqrt(128)

    float sc[Cn];
#pragma unroll
    for (int c = 0; c < Cn; ++c) {
        const float4 kv = *(const float4*)(k + ((size_t)bt * Cn + c) * En + hoff);
        float s = qv.x * kv.x + qv.y * kv.y + qv.z * kv.z + qv.w * kv.w;
#pragma unroll
        for (int m = 16; m >= 1; m >>= 1) s += __shfl_xor(s, m, 32);
        sc[c] = s * scale;
    }
    float mx = sc[0];
#pragma unroll
    for (int c = 1; c < Cn; ++c) mx = fmaxf(mx, sc[c]);
    float sum = 0.0f;
#pragma unroll
    for (int c = 0; c < Cn; ++c) { sc[c] = expf(sc[c] - mx); sum += sc[c]; }
    const float inv = 1.0f / sum;

    float4 o = make_float4(0.f, 0.f, 0.f, 0.f);
#pragma unroll
    for (int c = 0; c < Cn; ++c) {
        const float4 vv = *(const float4*)(v + ((size_t)bt * Cn + c) * En + hoff);
        const float a = sc[c] * inv;
        o.x += a * vv.x; o.y += a * vv.y; o.z += a * vv.z; o.w += a * vv.w;
    }
    *(float4*)(out + (size_t)bt * En + hoff) = o;
}

// ---------------- row-wise L2 normalize (wave per row) ----------------
__global__ __launch_bounds__(32)
void l2norm_rows(float* __restrict__ t, int cols)
{
    const int row = blockIdx.x;
    const int lane = threadIdx.x;
    float* p = t + (size_t)row * cols;
    float s = 0.0f;
    for (int j = lane; j < cols; j += 32) { const float x = p[j]; s += x * x; }
#pragma unroll
    for (int m = 16; m >= 1; m >>= 1) s += __shfl_xor(s, m, 32);
    const float inv = 1.0f / fmaxf(sqrtf(s), 1e-12f);
    for (int j = lane; j < cols; j += 32) p[j] *= inv;
}

// ---------------- final 256->2 head ----------------
__global__ void head_out(const float* __restrict__ Hd, const float* __restrict__ w2,
                         const float* __restrict__ b2, float* __restrict__ out, int rows)
{
    const int idx = blockIdx.x * blockDim.x + threadIdx.x;
    if (idx >= rows * 2) return;
    const int r = idx >> 1, c = idx & 1;
    const float* h = Hd + (size_t)r * 256;
    const float* w = w2 + (size_t)c * 256;
    float acc = b2[c];
    for (int j = 0; j < 256; ++j) acc += h[j] * w[j];
    out[idx] = acc;
}

// ---------------- host orchestration ----------------
extern "C" void kernel_launch(void* const* d_in, const int* in_sizes, int n_in,
                              void* d_out, int out_size, void* d_ws, size_t ws_size,
                              hipStream_t stream)
{
    (void)in_sizes; (void)n_in; (void)out_size; (void)ws_size;
    auto F = [&](int i) { return (const float*)d_in[i]; };
    const float* seq = F(0);
    const int* tids = (const int*)d_in[1];
    const int* cids = (const int*)d_in[2];

    // workspace layout (floats); big buffers sized for columns, reused for tables
    float* ws = (float*)d_ws;
    size_t off = 0;
    auto alloc = [&](size_t n) { float* p = ws + off; off += n; return p; };
    float* XG = alloc((size_t)6144 * 768);    // gathered spans
    float* XW = alloc((size_t)6144 * 2048);   // xw = X @ w_ih^T + b
    float* X1 = alloc((size_t)6144 * 1024);   // layer-1 input (hs_f | hs_b)
    float* GB = alloc((size_t)1024 * 2048);   // per-step gate preactivations
    float* Hb = alloc((size_t)1024 * 512);    // hidden state
    float* Cb = alloc((size_t)1024 * 512);    // cell state
    float* PR = alloc((size_t)1024 * 1024);   // [hf | hb] pre-projection
    float* BS = alloc(2048);                  // b_ih + b_hh
    float* TP = alloc((size_t)128 * 1024);    // tpool (post-proj)
    float* CP = alloc((size_t)1024 * 1024);   // cpool (post-proj)
    float* Qb = alloc((size_t)128 * 1024);
    float* Kb = alloc((size_t)1024 * 1024);
    float* Vb = alloc((size_t)1024 * 1024);
    float* AT = alloc((size_t)128 * 1024);    // attention output
    float* TR = alloc((size_t)128 * 1024);    // residual + normalized
    float* H1 = alloc((size_t)1024 * 256);    // head hidden

    auto gemm = [&](const float* A, int lda, const float* W, int ldw, const float* bias,
                    const float* Z, long long ldz, float* Cp, int ldc,
                    int M, int N, int K, int leaky) {
        wmma_gemm<<<dim3(N / 64, M / 128), 128, 0, stream>>>(A, lda, W, ldw, bias, Z, ldz,
                                                             Cp, ldc, K, leaky);
    };

    // One full 2-layer BiLSTM span-pool model (pbase = first param index of its LSTM)
    auto run_model = [&](int pbase, int Ns, int L, const int* ids, int rows_per_b,
                         const float* pw, const float* pb, float* poolOut) {
        gather_rows<<<Ns * L, 192, 0, stream>>>(seq, ids, XG, rows_per_b);
        for (int layer = 0; layer < 2; ++layer) {
            const int din = layer ? En : Dn;
            const float* Ain = layer ? X1 : XG;
            for (int dir = 0; dir < 2; ++dir) {
                const int wb = pbase + layer * 8 + dir * 4;  // w_ih, w_hh, b_ih, b_hh
                vec_add<<<8, 256, 0, stream>>>(F(wb + 2), F(wb + 3), BS, Gn);
                // xw for all (span, t): [Ns*L, din] @ [Gn, din]^T
                gemm(Ain, din, F(wb + 0), din, BS, nullptr, 0, XW, Gn, Ns * L, Gn, din, 0);
                hipMemsetAsync(Hb, 0, (size_t)Ns * Hn * sizeof(float), stream);
                hipMemsetAsync(Cb, 0, (size_t)Ns * Hn * sizeof(float), stream);
                for (int step = 0; step < L; ++step) {
                    const int tt = dir ? (L - 1 - step) : step;  // bwd = reversed time
                    // g = h @ w_hh^T + xw[:, tt, :]
                    gemm(Hb, Hn, F(wb + 1), Hn, nullptr,
                         XW + (size_t)tt * Gn, (long long)L * Gn, GB, Gn, Ns, Gn, Hn, 0);
                    float* hs = nullptr; long long hstr = 0;
                    if (layer == 0) {  // scatter h into layer-1 input at original time order
                        hs = X1 + (size_t)tt * En + dir * Hn;
                        hstr = (long long)L * En;
                    }
                    lstm_gate<<<(Ns * Hn + 255) / 256, 256, 0, stream>>>(GB, Cb, Hb, hs, hstr, Ns);
                }
                if (layer == 1)  // final hidden of last layer -> pool halves
                    copy_cols<<<(Ns * Hn + 255) / 256, 256, 0, stream>>>(Hb, PR + dir * Hn,
                                                                         Ns * Hn, En);
            }
        }
        // leaky(pool @ pool_w^T + pool_b)
        gemm(PR, En, pw, En, pb, nullptr, 0, poolOut, En, Ns, En, En, 1);
    };

    // columns first (big buffers), then tables (reuse)
    run_model(19, Bn * NCn, LCn, cids, NCn * LCn, F(37), F(38), CP);
    run_model(3,  Bn * Tn,  LTn, tids, Tn * LTn,  F(35), F(36), TP);

    // MHA: q from tables, k/v from columns (in_proj rows: [Wq; Wk; Wv])
    const float* Wi = F(39); const float* bi = F(40);
    gemm(TP, En, Wi,                       En, bi,        nullptr, 0, Qb, En, Bn * Tn,  En, En, 0);
    gemm(CP, En, Wi + (size_t)1 * En * En, En, bi + En,   nullptr, 0, Kb, En, Bn * NCn, En, En, 0);
    gemm(CP, En, Wi + (size_t)2 * En * En, En, bi + 2*En, nullptr, 0, Vb, En, Bn * NCn, En, En, 0);
    mha_attn<<<Bn * Tn * NH, 32, 0, stream>>>(Qb, Kb, Vb, AT);
    // out_proj with residual tpool, then row L2 normalize
    gemm(AT, En, F(41), En, F(42), TP, En, TR, En, Bn * Tn, En, En, 0);
    l2norm_rows<<<Bn * Tn, 32, 0, stream>>>(TR, En);

    float* out = (float*)d_out;
    // table head: leaky(TR @ w1^T + b1) @ w2^T + b2  -> out[0:256]
    gemm(TR, En, F(43), En, F(44), nullptr, 0, H1, 256, Bn * Tn, 256, En, 1);
    head_out<<<1, 256, 0, stream>>>(H1, F(45), F(46), out, Bn * Tn);
    // column head on cpool -> out[256:2304]
    gemm(CP, En, F(47), En, F(48), nullptr, 0, H1, 256, Bn * NCn, 256, En, 1);
    head_out<<<8, 256, 0, stream>>>(H1, F(49), F(50), out + 256, Bn * NCn);
}